// GIN_21191368639148
// MI455X (gfx1250) — compile-verified
//
#include <hip/hip_runtime.h>

// ---------------------------------------------------------------------------
// GIN forward (5 layers, D=128) for MI455X (gfx1250, wave32).
//  - edge aggregation: f32 global atomics (L2-resident working set)
//  - MLP GEMMs: V_WMMA_F32_16X16X4_F32, all operands in LDS.
//    Weights stored TRANSPOSED in LDS so every fragment (A and B) is one
//    8-byte ds_load_b64 into an even-aligned VGPR pair; stride 132 makes all
//    fragment reads hit 64 distinct banks.
//  - z-tile is staged via GLOBAL_LOAD_ASYNC_TO_LDS_B128 (ASYNCcnt), issued at
//    kernel entry and overlapped with the weight-transpose staging.
// ---------------------------------------------------------------------------

namespace {
constexpr int NN = 50000;   // nodes
constexpr int EE = 800000;  // edges
constexpr int DD = 128;     // feature dim
constexpr int LL = 5;       // layers
constexpr int GG = 1000;    // graphs
constexpr int TSTR = 132;   // LDS row stride (floats) for all tiles
// LDS layout (bytes from start of dynamic LDS):
constexpr int SZ_TILE = 128 * TSTR * 4;       // 67,584 B
constexpr int OFF_W1T = 0;                    // WT1 [n][k]
constexpr int OFF_W2T = SZ_TILE;              // WT2 [n][k]
constexpr int OFF_Z   = 2 * SZ_TILE;          // z tile [r][k]
constexpr int OFF_T   = 3 * SZ_TILE;          // intermediate [r][k]
}

typedef float v2f __attribute__((ext_vector_type(2)));
typedef float v8f __attribute__((ext_vector_type(8)));

__device__ __forceinline__ float atom_add_f32(float* p, float v) {
  return unsafeAtomicAdd(p, v);  // lowers to global_atomic_add_f32
}

__device__ __forceinline__ v8f wmma_f32(v2f a, v2f b, v8f c) {
  return __builtin_amdgcn_wmma_f32_16x16x4_f32(false, a, false, b, (short)0, c,
                                               false, false);
}

// ---------------------------------------------------------------- utilities
__global__ __launch_bounds__(256) void copy_f4(const float4* __restrict__ in,
                                               float4* __restrict__ out, int n4) {
  int i = blockIdx.x * blockDim.x + threadIdx.x;
  if (i < n4) out[i] = in[i];
}

__global__ __launch_bounds__(256) void zero_f(float* __restrict__ p, int n) {
  int i = blockIdx.x * blockDim.x + threadIdx.x;
  if (i < n) p[i] = 0.f;
}

// ------------------------------------------------ edge scatter: z += h[src]
__global__ __launch_bounds__(256) void edge_scatter(const float* __restrict__ feat,
                                                    const int* __restrict__ esrc,
                                                    const int* __restrict__ edst,
                                                    float* __restrict__ z) {
  int gt = blockIdx.x * blockDim.x + threadIdx.x;
  int e = gt >> 5;
  if (e >= EE) return;
  int lane = gt & 31;
  int s = esrc[e];
  int d = edst[e];
  const float4 v = *(const float4*)(feat + (size_t)s * DD + (lane << 2));
  float* zp = z + (size_t)d * DD + (lane << 2);
  atom_add_f32(zp + 0, v.x);
  atom_add_f32(zp + 1, v.y);
  atom_add_f32(zp + 2, v.z);
  atom_add_f32(zp + 3, v.w);
}

// -------------------------------------------------------------- fused MLP
// out = relu( relu(z @ W1 + b1) @ W2 + b2 ), 128-row tile per block.
// 8 waves; wave w -> wrow = w>>1 (2 row-tiles), wcol = w&1 (4 col-tiles):
// a 32x64 output patch, 8 v8f accumulators.
//
// WMMA f32 16x16x4 fragment layouts (ISA 7.12.2), half = lane>>4:
//   A (16x4): lane = M (mod 16); VGPR0/1 = K = 2*half + {0,1}
//   B (4x16): lane = N (mod 16); VGPR0/1 = K = 2*half + {0,1}
//     -> with transposed weights WT[n][k], a B frag is v2f at WT[n] + kb
//   C/D    : VGPR i = row (i + 8*half), col = lane&15
__global__ __launch_bounds__(256) void gin_mlp(const float* __restrict__ z,
                                               const float* __restrict__ W1,
                                               const float* __restrict__ b1,
                                               const float* __restrict__ W2,
                                               const float* __restrict__ b2,
                                               float* __restrict__ out) {
  extern __shared__ float lds[];
  float* lw1t = lds + OFF_W1T / 4;  // [128][TSTR]  WT1
  float* lw2t = lds + OFF_W2T / 4;  // [128][TSTR]  WT2
  float* lz   = lds + OFF_Z / 4;    // [128][TSTR]  z tile
  float* lt   = lds + OFF_T / 4;    // [128][TSTR]  relu(z@W1+b1)

  const int tid   = threadIdx.x;
  const int lane  = tid & 31;
  const int half  = lane >> 4;   // 0 | 1
  const int l16   = lane & 15;
  const int w     = tid >> 5;    // wave 0..7
  const int wrow  = w >> 1;      // 0..3  -> rows [wrow*32, wrow*32+32)
  const int wcol  = w & 1;       // 0..1  -> cols [wcol*64, wcol*64+64)
  const int mbase = blockIdx.x * 128;

  // ---- async DMA: z tile -> LDS (overlapped with weight staging below) ---
  const unsigned ldsZ = __builtin_amdgcn_groupstaticsize() + (unsigned)OFF_Z;
#pragma unroll
  for (int i = 0; i < 16; ++i) {
    const int idx4 = tid + i * 256;      // 0..4095 float4 slots
    const int r    = idx4 >> 5;          // tile row 0..127
    const int c    = (idx4 & 31) << 2;   // col 0..124 step 4
    int zr = mbase + r;
    if (zr > NN - 1) zr = NN - 1;        // clamp; OOB rows never stored
    const unsigned la = ldsZ + (unsigned)(r * TSTR + c) * 4u;
    const unsigned ga = ((unsigned)zr * (unsigned)DD + (unsigned)c) * 4u;
    asm volatile("global_load_async_to_lds_b128 %0, %1, %2"
                 :: "v"(la), "v"(ga), "s"(z)
                 : "memory");
  }

  // ---- synchronous staging: W1^T, W2^T -> LDS ----------------------------
  // lane = source column (coalesced global reads), 4 source rows per float4.
#pragma unroll
  for (int i = 0; i < 16; ++i) {
    const int s  = tid + i * 256;        // 0..4095
    const int c  = s & 127;              // source column = WT row
    const int r0 = (s >> 7) << 2;        // source row group (4 rows)
    float4 v1, v2;
    v1.x = W1[(size_t)(r0 + 0) * DD + c];
    v1.y = W1[(size_t)(r0 + 1) * DD + c];
    v1.z = W1[(size_t)(r0 + 2) * DD + c];
    v1.w = W1[(size_t)(r0 + 3) * DD + c];
    v2.x = W2[(size_t)(r0 + 0) * DD + c];
    v2.y = W2[(size_t)(r0 + 1) * DD + c];
    v2.z = W2[(size_t)(r0 + 2) * DD + c];
    v2.w = W2[(size_t)(r0 + 3) * DD + c];
    *(float4*)(lw1t + c * TSTR + r0) = v1;
    *(float4*)(lw2t + c * TSTR + r0) = v2;
  }

  asm volatile("s_wait_asynccnt 0" ::: "memory");  // z tile DMA complete (this wave)
  __syncthreads();                                 // all waves' tiles visible

  // ---- GEMM1: acc = z_tile @ W1 ------------------------------------------
  const float* a0p = lz + (wrow * 32 + l16) * TSTR;
  const float* a1p = a0p + 16 * TSTR;
  const float* bp1[4];
#pragma unroll
  for (int j = 0; j < 4; ++j) bp1[j] = lw1t + ((wcol * 4 + j) * 16 + l16) * TSTR;

  v8f acc[2][4] = {};
#pragma unroll 4
  for (int k0 = 0; k0 < DD; k0 += 4) {
    const int kb = k0 + half * 2;
    v2f a0 = *(const v2f*)(a0p + kb);    // ds_load_b64, 8B aligned
    v2f a1 = *(const v2f*)(a1p + kb);
#pragma unroll
    for (int j = 0; j < 4; ++j) {
      v2f b = *(const v2f*)(bp1[j] + kb);  // ds_load_b64 (transposed weights)
      acc[0][j] = wmma_f32(a0, b, acc[0][j]);
      acc[1][j] = wmma_f32(a1, b, acc[1][j]);
    }
  }

  // ---- bias + relu -> lt (re-layout for GEMM2 A-fragments) ---------------
#pragma unroll
  for (int rt = 0; rt < 2; ++rt) {
#pragma unroll
    for (int j = 0; j < 4; ++j) {
      const int n = (wcol * 4 + j) * 16 + l16;
      const float bias = b1[n];
#pragma unroll
      for (int i = 0; i < 8; ++i) {
        float v = acc[rt][j][i] + bias;
        v = v > 0.f ? v : 0.f;
        const int r = wrow * 32 + rt * 16 + i + half * 8;
        lt[r * TSTR + n] = v;
      }
    }
  }
  __syncthreads();  // cross-wave: column halves of lt come from partner wave

  // ---- GEMM2: acc2 = lt @ W2 ---------------------------------------------
  const float* t0p = lt + (wrow * 32 + l16) * TSTR;
  const float* t1p = t0p + 16 * TSTR;
  const float* bp2[4];
#pragma unroll
  for (int j = 0; j < 4; ++j) bp2[j] = lw2t + ((wcol * 4 + j) * 16 + l16) * TSTR;

  v8f acc2[2][4] = {};
#pragma unroll 4
  for (int k0 = 0; k0 < DD; k0 += 4) {
    const int kb = k0 + half * 2;
    v2f a0 = *(const v2f*)(t0p + kb);
    v2f a1 = *(const v2f*)(t1p + kb);
#pragma unroll
    for (int j = 0; j < 4; ++j) {
      v2f b = *(const v2f*)(bp2[j] + kb);
      acc2[0][j] = wmma_f32(a0, b, acc2[0][j]);
      acc2[1][j] = wmma_f32(a1, b, acc2[1][j]);
    }
  }

  // ---- bias + relu -> global ---------------------------------------------
#pragma unroll
  for (int rt = 0; rt < 2; ++rt) {
#pragma unroll
    for (int j = 0; j < 4; ++j) {
      const int n = (wcol * 4 + j) * 16 + l16;
      const float bias = b2[n];
#pragma unroll
      for (int i = 0; i < 8; ++i) {
        const int row = mbase + wrow * 32 + rt * 16 + i + half * 8;
        if (row < NN) {
          float v = acc2[rt][j][i] + bias;
          v = v > 0.f ? v : 0.f;
          out[(size_t)row * DD + n] = v;
        }
      }
    }
  }
}

// -------------------------------------------------- global mean pool (sums)
__global__ __launch_bounds__(256) void pool_scatter(const float* __restrict__ h,
                                                    const int* __restrict__ batch,
                                                    float* __restrict__ pooled,
                                                    float* __restrict__ cnt) {
  int gt = blockIdx.x * blockDim.x + threadIdx.x;
  int node = gt >> 5;
  if (node >= NN) return;
  int lane = gt & 31;
  int g = batch[node];
  const float4 v = *(const float4*)(h + (size_t)node * DD + (lane << 2));
  float* pp = pooled + (size_t)g * DD + (lane << 2);
  atom_add_f32(pp + 0, v.x);
  atom_add_f32(pp + 1, v.y);
  atom_add_f32(pp + 2, v.z);
  atom_add_f32(pp + 3, v.w);
  if (lane == 0) atom_add_f32(cnt + g, 1.f);
}

// ----------------------------------------------------- head: (pooled/cnt)@Wh
__global__ __launch_bounds__(256) void head(const float* __restrict__ pooled,
                                            const float* __restrict__ cnt,
                                            const float* __restrict__ Wh,
                                            const float* __restrict__ bh,
                                            float* __restrict__ out) {
  int g = blockIdx.x * 8 + (threadIdx.x >> 5);
  if (g >= GG) return;
  int lane = threadIdx.x & 31;
  float s = 0.f;
#pragma unroll
  for (int j = lane; j < DD; j += 32) s += pooled[(size_t)g * DD + j] * Wh[j];
#pragma unroll
  for (int off = 16; off > 0; off >>= 1) s += __shfl_xor(s, off, 32);
  if (lane == 0) {
    float c = cnt[g];
    c = c > 1.f ? c : 1.f;
    out[g] = s / c + bh[0];
  }
}

// ---------------------------------------------------------------------------
extern "C" void kernel_launch(void* const* d_in, const int* in_sizes, int n_in,
                              void* d_out, int out_size, void* d_ws, size_t ws_size,
                              hipStream_t stream) {
  const float* x   = (const float*)d_in[0];
  const int* eidx  = (const int*)d_in[1];    // [2,E]: row0=src, row1=dst
  const int* batch = (const int*)d_in[2];
  const float* W1s = (const float*)d_in[3];  // [L,D,D]
  const float* b1s = (const float*)d_in[4];  // [L,D]
  const float* W2s = (const float*)d_in[5];  // [L,D,D]
  const float* b2s = (const float*)d_in[6];  // [L,D]
  const float* Wh  = (const float*)d_in[7];  // [D,1]
  const float* bh  = (const float*)d_in[8];  // [1]
  float* out = (float*)d_out;                // [G,1]

  // workspace layout: h (N*D) | z (N*D) | pooled (G*D) | cnt (G)  ~51.7 MB
  float* hbuf   = (float*)d_ws;
  float* zbuf   = hbuf + (size_t)NN * DD;
  float* pooled = zbuf + (size_t)NN * DD;
  float* cnt    = pooled + (size_t)GG * DD;

  const int* esrc = eidx;
  const int* edst = eidx + EE;

  const int n4 = NN * DD / 4;
  const int copyBlocks = (n4 + 255) / 256;
  const int scatBlocks = (EE * 32) / 256;
  const int mlpBlocks  = (NN + 127) / 128;
  const int poolBlocks = (NN * 32 + 255) / 256;
  const size_t ldsBytes = (size_t)4 * SZ_TILE;  // 270,336 B

  for (int l = 0; l < LL; ++l) {
    const float* feat = (l == 0) ? x : hbuf;
    copy_f4<<<copyBlocks, 256, 0, stream>>>((const float4*)feat, (float4*)zbuf, n4);
    edge_scatter<<<scatBlocks, 256, 0, stream>>>(feat, esrc, edst, zbuf);
    gin_mlp<<<mlpBlocks, 256, ldsBytes, stream>>>(
        zbuf, W1s + (size_t)l * DD * DD, b1s + (size_t)l * DD,
        W2s + (size_t)l * DD * DD, b2s + (size_t)l * DD, hbuf);
  }

  zero_f<<<(GG * DD + GG + 255) / 256, 256, 0, stream>>>(pooled, GG * DD + GG);
  pool_scatter<<<poolBlocks, 256, 0, stream>>>(hbuf, batch, pooled, cnt);
  head<<<(GG + 7) / 8, 256, 0, stream>>>(pooled, cnt, Wh, bh, out);
}